// FourierLayer_59889023976286
// MI455X (gfx1250) — compile-verified
//
#include <hip/hip_runtime.h>
#include <hip/hip_bf16.h>

typedef __attribute__((ext_vector_type(2))) float v2f;
typedef __attribute__((ext_vector_type(4))) float v4f;
typedef __attribute__((ext_vector_type(8))) float v8f;

#define BATCH 8
#define HDIM 256
#define WDIM 256
#define CDIM 128
#define KCF  65     // C/2+1
#define KCP  80     // padded to 5 tiles of 16
#define NKW  32     // surviving W-frequencies (16 low + 16 high)
#define ROWF (WDIM*CDIM)  // 32768 floats per (b,h) row-block

// workspace offsets in floats
#define OFF_GC  0         // [32][256] cos(2pi kw w /256)
#define OFF_GS  8192      // [32][256] sin
#define OFF_FC  16384     // [128][80] cos(2pi kc c /128), 0 for kc>=65
#define OFF_FS  26624     // [128][80] sin
#define OFF_A   36864     // [80][128] alpha*cos/128
#define OFF_B   47104     // [80][128] -alpha*sin/128
#define OFF_HC  57344     // [256][32] cos/256
#define OFF_HS  65536     // [256][32] sin/256
#define OFF_WRE 73728     // [32][32][80]
#define OFF_WIM 155648
#define OFF_BRE 237568
#define OFF_BIM 319488

__device__ __forceinline__ v8f z8() {
  v8f z = {0.f,0.f,0.f,0.f,0.f,0.f,0.f,0.f};
  return z;
}

// D = A(16x4) * B(4x16) + C, fp32 in / fp32 acc
__device__ __forceinline__ v8f wmma4(v2f a, v2f b, v8f c) {
  return __builtin_amdgcn_wmma_f32_16x16x4_f32(
      /*neg_a=*/false, a, /*neg_b=*/false, b,
      /*c_mod=*/(short)0, c, /*reuse_a=*/false, /*reuse_b=*/false);
}

// A-fragment: 16x4 tile at (row0,k0), leading dim lda.
// lane 0..15: M=lane, K=k0..k0+1 ; lane 16..31: M=lane-16, K=k0+2..k0+3
__device__ __forceinline__ v2f ldA(const float* Ab, int lda, int row0, int k0, int lane) {
  int l = lane & 15, hf = lane >> 4;
  const float* p = Ab + (size_t)(row0 + l) * lda + k0 + 2 * hf;
  v2f r = {p[0], p[1]};
  return r;
}

// B-fragment: 4x16 tile at (k0,col0), leading dim ldb.
// lane 0..15: N=lane, K=k0..k0+1 ; lane 16..31: N=lane-16, K=k0+2..k0+3
__device__ __forceinline__ v2f ldB(const float* Bb, int ldb, int k0, int col0, int lane) {
  int l = lane & 15, hf = lane >> 4;
  const float* p = Bb + (size_t)(k0 + 2 * hf) * ldb + col0 + l;
  v2f r = {p[0], p[ldb]};
  return r;
}

// C/D: 16x16 tile at (m0,n0). VGPR r: lanes0-15 -> M=m0+r, lanes16-31 -> M=m0+r+8
__device__ __forceinline__ void stC(float* Cb, int ldc, int m0, int n0, int lane, v8f c) {
  int l = lane & 15, hf = lane >> 4;
  for (int r = 0; r < 8; ++r)
    Cb[(size_t)(m0 + r + 8 * hf) * ldc + n0 + l] = c[r];
}

// Non-temporal C store (write-once global output stream)
__device__ __forceinline__ void stC_nt(float* Cb, int ldc, int m0, int n0, int lane, v8f c) {
  int l = lane & 15, hf = lane >> 4;
  for (int r = 0; r < 8; ++r)
    __builtin_nontemporal_store(c[r], Cb + (size_t)(m0 + r + 8 * hf) * ldc + n0 + l);
}

// ---- forward stages 1+2: X(256x128) -> Z(32x80 complex) at the 32 corner kw ----
// Vre = Gc*X, Vim = -(Gs*X)  (e^{-i th})
// Zre = Vre*Fc + Vim*Fs ; Zim = Vim*Fc - Vre*Fs
__device__ void fwd_stage12(const float* __restrict__ X, const float* __restrict__ ws,
                            float* sVre, float* sVim,
                            float* Zre, float* Zim, int tid) {
  const int wid = tid >> 5, lane = tid & 31;
  const float* Gc = ws + OFF_GC;
  const float* Gs = ws + OFF_GS;
  const float* Fc = ws + OFF_FC;
  const float* Fs = ws + OFF_FS;
  // stage 1: M=32, N=128, K=256. One n-tile per wave; both m-tiles (re+im) in
  // registers so each X (B) fragment is loaded exactly once from HBM.
  {
    int n0 = wid * 16;
    v8f aR0 = z8(), aR1 = z8(), aI0 = z8(), aI1 = z8();
    for (int k0 = 0; k0 < 256; k0 += 4) {
      v2f bx  = ldB(X, CDIM, k0, n0, lane);
      v2f ac0 = ldA(Gc, 256, 0,  k0, lane);
      v2f ac1 = ldA(Gc, 256, 16, k0, lane);
      v2f as0 = ldA(Gs, 256, 0,  k0, lane);
      v2f as1 = ldA(Gs, 256, 16, k0, lane);
      aR0 = wmma4(ac0, bx, aR0);
      aR1 = wmma4(ac1, bx, aR1);
      aI0 = wmma4(as0, bx, aI0);
      aI1 = wmma4(as1, bx, aI1);
    }
    stC(sVre, CDIM, 0,  n0, lane, aR0);
    stC(sVre, CDIM, 16, n0, lane, aR1);
    v8f n0v, n1v;
    for (int r = 0; r < 8; ++r) { n0v[r] = -aI0[r]; n1v[r] = -aI1[r]; }
    stC(sVim, CDIM, 0,  n0, lane, n0v);
    stC(sVim, CDIM, 16, n0, lane, n1v);
  }
  __syncthreads();
  // stage 2: M=32, N=80, K=128  (B = Fc/Fs, L2-resident; A from LDS)
  for (int t = wid; t < 10; t += 8) {
    int m0 = (t / 5) * 16, n0 = (t % 5) * 16;
    v8f zr = z8(), za = z8(), zb = z8();
    for (int k0 = 0; k0 < 128; k0 += 4) {
      v2f ar = ldA(sVre, CDIM, m0, k0, lane);
      v2f ai = ldA(sVim, CDIM, m0, k0, lane);
      v2f bc = ldB(Fc, KCP, k0, n0, lane);
      v2f bs = ldB(Fs, KCP, k0, n0, lane);
      zr = wmma4(ar, bc, zr);
      zr = wmma4(ai, bs, zr);
      za = wmma4(ai, bc, za);   // Vim*Fc
      zb = wmma4(ar, bs, zb);   // Vre*Fs
    }
    stC(Zre, KCP, m0, n0, lane, zr);
    v8f zi;
    for (int r = 0; r < 8; ++r) zi[r] = za[r] - zb[r];
    stC(Zim, KCP, m0, n0, lane, zi);
  }
  __syncthreads();
}

// ---- K1: fill DFT bases ----
__global__ void k_bases(float* __restrict__ ws) {
  int i = blockIdx.x * blockDim.x + threadIdx.x;
  if (i < 32 * 256) {            // Gc/Gs [j][w]
    int j = i >> 8, w = i & 255;
    int kw = j < 16 ? j : j + 224;
    float t = 2.0f * (float)((kw * w) & 255) / 256.0f;
    ws[OFF_GC + i] = cospif(t);
    ws[OFF_GS + i] = sinpif(t);
  }
  if (i < 128 * 80) {            // Fc/Fs [c][kc]
    int c = i / 80, kc = i % 80;
    float cv = 0.f, sv = 0.f;
    if (kc < KCF) {
      float t = 2.0f * (float)((kc * c) & 127) / 128.0f;
      cv = cospif(t); sv = sinpif(t);
    }
    ws[OFF_FC + i] = cv;
    ws[OFF_FS + i] = sv;
  }
  if (i < 80 * 128) {            // A/Bm [kc][c]
    int kc = i / 128, c = i % 128;
    float av = 0.f, bv = 0.f;
    if (kc < KCF) {
      float alpha = (kc == 0 || kc == 64) ? 1.0f : 2.0f;
      float t = 2.0f * (float)((kc * c) & 127) / 128.0f;
      av =  alpha * cospif(t) * (1.0f / 128.0f);
      bv = -alpha * sinpif(t) * (1.0f / 128.0f);
    }
    ws[OFF_A + i] = av;
    ws[OFF_B + i] = bv;
  }
  if (i < 256 * 32) {            // Hc/Hs [w][j]  (e^{+i th} / 256)
    int w = i / 32, j = i % 32;
    int kw = j < 16 ? j : j + 224;
    float t = 2.0f * (float)((kw * w) & 255) / 256.0f;
    ws[OFF_HC + i] = cospif(t) * (1.0f / 256.0f);
    ws[OFF_HS + i] = sinpif(t) * (1.0f / 256.0f);
  }
}

// ---- K2: Wft/Bft for the 32 corner h rows ----
__global__ void __launch_bounds__(256) k_wft(const float* __restrict__ w,
                                             const float* __restrict__ b,
                                             float* __restrict__ ws) {
  __shared__ float sV[2][NKW * CDIM];
  int j = blockIdx.x;
  int h = j < 16 ? j : j + 224;
  int tid = threadIdx.x;
  fwd_stage12(w + (size_t)h * ROWF, ws, sV[0], sV[1],
              ws + OFF_WRE + j * (NKW * KCP), ws + OFF_WIM + j * (NKW * KCP), tid);
  fwd_stage12(b + (size_t)h * ROWF, ws, sV[0], sV[1],
              ws + OFF_BRE + j * (NKW * KCP), ws + OFF_BIM + j * (NKW * KCP), tid);
}

// ---- K3: main per-(batch, corner-h) pipeline ----
__global__ void __launch_bounds__(256) k_main(const float* __restrict__ x,
                                              const float* __restrict__ ws,
                                              float* __restrict__ out) {
  __shared__ float sV[2][NKW * CDIM];   // V, later R1/R2  (32 KB)
  __shared__ float sZ[2][NKW * KCP];    // Z, in-place P   (20 KB)
  int tid = threadIdx.x;
  int bb = blockIdx.x >> 5;
  int j  = blockIdx.x & 31;
  int h  = j < 16 ? j : j + 224;

  fwd_stage12(x + ((size_t)bb * HDIM + h) * ROWF, ws, sV[0], sV[1], sZ[0], sZ[1], tid);

  // stage 3: P = Z * Wft[h] + Bft[h]  (complex, in place; zero the pad columns)
  const float* Wre = ws + OFF_WRE + j * (NKW * KCP);
  const float* Wim = ws + OFF_WIM + j * (NKW * KCP);
  const float* Bre = ws + OFF_BRE + j * (NKW * KCP);
  const float* Bim = ws + OFF_BIM + j * (NKW * KCP);
  for (int idx = tid; idx < NKW * KCP; idx += 256) {
    int kc = idx % KCP;
    float pr = 0.f, pi = 0.f;
    if (kc < KCF) {
      float zr = sZ[0][idx], zi = sZ[1][idx];
      float wr = Wre[idx], wi = Wim[idx];
      pr = zr * wr - zi * wi + Bre[idx];
      pi = zr * wi + zi * wr + Bim[idx];
    }
    sZ[0][idx] = pr;
    sZ[1][idx] = pi;
  }
  __syncthreads();

  // stage 4: R1 = Pre*A + Pim*Bm ; R2 = Pre*Bm - Pim*A   (M=32,N=128,K=80)
  const float* Am = ws + OFF_A;
  const float* Bm = ws + OFF_B;
  int wid = tid >> 5, lane = tid & 31;
  for (int t = wid; t < 16; t += 8) {
    int m0 = (t >> 3) * 16, n0 = (t & 7) * 16;
    v8f r1 = z8(), ra = z8(), rb = z8();
    for (int k0 = 0; k0 < KCP; k0 += 4) {
      v2f ar = ldA(sZ[0], KCP, m0, k0, lane);
      v2f ai = ldA(sZ[1], KCP, m0, k0, lane);
      v2f ba = ldB(Am, CDIM, k0, n0, lane);
      v2f bb = ldB(Bm, CDIM, k0, n0, lane);
      r1 = wmma4(ar, ba, r1);
      r1 = wmma4(ai, bb, r1);
      ra = wmma4(ar, bb, ra);   // Pre*Bm
      rb = wmma4(ai, ba, rb);   // Pim*A
    }
    stC(sV[0], CDIM, m0, n0, lane, r1);
    v8f r2;
    for (int r = 0; r < 8; ++r) r2[r] = ra[r] - rb[r];
    stC(sV[1], CDIM, m0, n0, lane, r2);
  }
  __syncthreads();

  // stage 5: out = Hc*R1 + Hs*R2   (M=256,N=128,K=32) -> global (non-temporal)
  const float* Hc = ws + OFF_HC;
  const float* Hs = ws + OFF_HS;
  float* O = out + ((size_t)bb * HDIM + h) * ROWF;
  for (int t = wid; t < 128; t += 8) {
    int m0 = (t >> 3) * 16, n0 = (t & 7) * 16;
    v8f acc = z8();
    for (int k0 = 0; k0 < 32; k0 += 4) {
      v2f ac = ldA(Hc, 32, m0, k0, lane);
      v2f as = ldA(Hs, 32, m0, k0, lane);
      v2f b1 = ldB(sV[0], CDIM, k0, n0, lane);
      v2f b2 = ldB(sV[1], CDIM, k0, n0, lane);
      acc = wmma4(ac, b1, acc);
      acc = wmma4(as, b2, acc);
    }
    stC_nt(O, CDIM, m0, n0, lane, acc);
  }
}

// ---- K4: zero-fill masked rows h in [16,240) (non-temporal stream) ----
__global__ void k_zero(float* __restrict__ out) {
  const int nvec = 224 * ROWF / 4;  // v4f per batch region
  v4f* o = (v4f*)(out + ((size_t)blockIdx.y * HDIM + 16) * ROWF);
  v4f z = {0.f, 0.f, 0.f, 0.f};
  for (int i = blockIdx.x * blockDim.x + threadIdx.x; i < nvec;
       i += gridDim.x * blockDim.x)
    __builtin_nontemporal_store(z, o + i);
}

extern "C" void kernel_launch(void* const* d_in, const int* in_sizes, int n_in,
                              void* d_out, int out_size, void* d_ws, size_t ws_size,
                              hipStream_t stream) {
  (void)in_sizes; (void)n_in; (void)out_size; (void)ws_size;
  const float* x = (const float*)d_in[0];
  const float* w = (const float*)d_in[1];
  const float* b = (const float*)d_in[2];
  float* out = (float*)d_out;
  float* ws  = (float*)d_ws;

  k_zero <<<dim3(2048, BATCH), 256, 0, stream>>>(out);
  k_bases<<<40, 256, 0, stream>>>(ws);
  k_wft  <<<32, 256, 0, stream>>>(w, b, ws);
  k_main <<<BATCH * NKW, 256, 0, stream>>>(x, ws, out);
}